// Attn_44470091383039
// MI455X (gfx1250) — compile-verified
//
#include <hip/hip_runtime.h>
#include <hip/hip_bf16.h>

// ---------------------------------------------------------------------------
// GAT-style edge-softmax attention for MI455X (gfx1250, wave32, WMMA).
// N=M=50000, E=640000, DIM=128, H=8, DH=16.
// GEMMs use V_WMMA_F32_16X16X4_F32; irregular edge work uses gathered 16-float
// head slices + float atomics so no [E,128] tensor ever touches HBM.
// ---------------------------------------------------------------------------

typedef __attribute__((ext_vector_type(2))) float v2f;
typedef __attribute__((ext_vector_type(8))) float v8f;

#define DIMC 128
#define HC   8
#define DHC  16

// ---------------------------------------------------------------------------
// Y[nrows,128] = X[nrows,128] @ W[128,128] + bias.
// Block = 256 threads = 8 waves. Block owns a 16-row stripe; wave w owns
// column tile [16w, 16w+16). A-tile (16x128) staged in LDS once per block.
// K-loop: 32 x V_WMMA_F32_16X16X4_F32.
// Fragment layouts (wave32):
//   A 16x4 : lanes 0-15 -> M=lane, {k,k+1}; lanes 16-31 -> M=lane-16, {k+2,k+3}
//   B 4x16 : lanes 0-15 -> N=lane, {k,k+1}; lanes 16-31 -> N=lane-16, {k+2,k+3}
//   C 16x16: VGPR r, lanes 0-15 -> M=r, N=lane; lanes 16-31 -> M=r+8, N=lane-16
// ---------------------------------------------------------------------------
__global__ __launch_bounds__(256) void gat_gemm128_kernel(
    const float* __restrict__ X, const float* __restrict__ W,
    const float* __restrict__ bias, float* __restrict__ Y, int nrows)
{
    __shared__ float As[16 * DIMC];

    const int row0 = blockIdx.x * 16;
    for (int i = threadIdx.x; i < 16 * DIMC; i += 256) {
        const int r = i >> 7, c = i & 127;
        const int gr = row0 + r;
        As[i] = (gr < nrows) ? X[gr * DIMC + c] : 0.0f;
    }
    __syncthreads();

    const int wave  = threadIdx.x >> 5;       // 0..7 -> column tile
    const int lane  = threadIdx.x & 31;
    const int col0  = wave * 16;
    const int nn    = lane & 15;              // N (B/C) and M (A) sub-index
    const int khalf = (lane >> 4) << 1;       // 0 for lanes 0-15, 2 for 16-31

    v8f acc = {};
    for (int k = 0; k < DIMC; k += 4) {
        v2f a, b;
        a.x = As[nn * DIMC + k + khalf + 0];
        a.y = As[nn * DIMC + k + khalf + 1];
        b.x = W[(k + khalf + 0) * DIMC + col0 + nn];
        b.y = W[(k + khalf + 1) * DIMC + col0 + nn];
        acc = __builtin_amdgcn_wmma_f32_16x16x4_f32(
            /*neg_a=*/false, a, /*neg_b=*/false, b,
            /*c_mod=*/(short)0, acc, /*reuse_a=*/false, /*reuse_b=*/false);
    }

    const int mbase = (lane >> 4) * 8;
    const float bv  = bias[col0 + nn];
    #pragma unroll
    for (int r = 0; r < 8; ++r) {
        const int gr = row0 + mbase + r;
        if (gr < nrows) Y[gr * DIMC + col0 + nn] = acc[r] + bv;
    }
}

__device__ __forceinline__ void atomicMaxFloat(float* addr, float val)
{
    // Valid because buffer is pre-initialized with a real float (attn_qq).
    if (val >= 0.0f) atomicMax((int*)addr, __float_as_int(val));
    else             atomicMin((unsigned int*)addr, __float_as_uint(val));
}

// attn_qq[n,h] = sum_d a[h,d]*prelu(Qq[n,h,d]+Kq[n,h,d]); maxv init.
__global__ __launch_bounds__(256) void gat_node_attn_kernel(
    const float* __restrict__ Qq, const float* __restrict__ Kq,
    const float* __restrict__ a, const float* __restrict__ alpha_p,
    float* __restrict__ attn_qq, float* __restrict__ maxv, int n)
{
    const int idx = blockIdx.x * 256 + threadIdx.x;   // node*H + h
    if (idx >= n * HC) return;
    const int node = idx >> 3, h = idx & 7;
    const float alpha = alpha_p[0];
    const int base = node * DIMC + h * DHC;
    float s = 0.0f;
    #pragma unroll
    for (int d = 0; d < DHC; ++d) {
        float x = Qq[base + d] + Kq[base + d];
        x = (x >= 0.0f) ? x : alpha * x;
        s = fmaf(a[h * DHC + d], x, s);
    }
    attn_qq[idx] = s;
    maxv[idx]    = s;
}

// attn_e[e,h] + scatter-max into maxv[qi,h].
__global__ __launch_bounds__(256) void gat_edge_attn_kernel(
    const float* __restrict__ Qq, const float* __restrict__ Kk,
    const int* __restrict__ qidx, const int* __restrict__ kidx,
    const float* __restrict__ a, const float* __restrict__ alpha_p,
    float* __restrict__ attn_e, float* __restrict__ maxv, int E)
{
    const int idx = blockIdx.x * 256 + threadIdx.x;   // edge*H + h
    if (idx >= E * HC) return;
    const int e = idx >> 3, h = idx & 7;
    const int qi = qidx[e], ki = kidx[e];
    const float alpha = alpha_p[0];
    const int qb = qi * DIMC + h * DHC;
    const int kb = ki * DIMC + h * DHC;
    float s = 0.0f;
    #pragma unroll
    for (int d = 0; d < DHC; ++d) {
        float x = Qq[qb + d] + Kk[kb + d];
        x = (x >= 0.0f) ? x : alpha * x;
        s = fmaf(a[h * DHC + d], x, s);
    }
    attn_e[idx] = s;
    atomicMaxFloat(&maxv[qi * HC + h], s);
}

// denom = exp_qq ; msg = exp_qq * Vq
__global__ __launch_bounds__(256) void gat_softmax_init_kernel(
    const float* __restrict__ attn_qq, const float* __restrict__ maxv,
    const float* __restrict__ Vq, float* __restrict__ denom,
    float* __restrict__ msg, int n)
{
    const int idx = blockIdx.x * 256 + threadIdx.x;   // node*H + h
    if (idx >= n * HC) return;
    const int node = idx >> 3, h = idx & 7;
    const float eq = __expf(attn_qq[idx] - maxv[idx]);
    denom[idx] = eq;
    const int base = node * DIMC + h * DHC;
    #pragma unroll
    for (int d = 0; d < DHC; ++d)
        msg[base + d] = eq * Vq[base + d];
}

// w = exp(attn_e - max[qi]); denom[qi,h] += w; msg[qi,h,:] += w * Vv[ki,h,:]
__global__ __launch_bounds__(256) void gat_edge_scatter_kernel(
    const float* __restrict__ attn_e, const float* __restrict__ maxv,
    const float* __restrict__ Vv, const int* __restrict__ qidx,
    const int* __restrict__ kidx, float* __restrict__ denom,
    float* __restrict__ msg, int E)
{
    const int idx = blockIdx.x * 256 + threadIdx.x;   // edge*H + h
    if (idx >= E * HC) return;
    const int e = idx >> 3, h = idx & 7;
    const int qi = qidx[e], ki = kidx[e];
    const float w = __expf(attn_e[idx] - maxv[qi * HC + h]);
    atomicAdd(&denom[qi * HC + h], w);
    const int qb = qi * DIMC + h * DHC;
    const int kb = ki * DIMC + h * DHC;
    #pragma unroll
    for (int d = 0; d < DHC; ++d)
        atomicAdd(&msg[qb + d], w * Vv[kb + d]);
}

// msg /= denom (in place)
__global__ __launch_bounds__(256) void gat_div_kernel(
    float* __restrict__ msg, const float* __restrict__ denom, int n)
{
    const int idx = blockIdx.x * 256 + threadIdx.x;   // node*128 + c
    if (idx >= n * DIMC) return;
    const int node = idx >> 7, h = (idx >> 4) & 7;
    float d = denom[node * HC + h];
    d = (d == 0.0f) ? 1.0f : d;
    msg[idx] = msg[idx] / d;
}

// res = query + P ; LayerNorm over 128. One wave32 per row, shfl reduction.
__global__ __launch_bounds__(256) void gat_residual_ln_kernel(
    const float* __restrict__ q, const float* __restrict__ P,
    const float* __restrict__ gamma, const float* __restrict__ beta,
    float* __restrict__ out, int n)
{
    const int wave = threadIdx.x >> 5;
    const int lane = threadIdx.x & 31;
    const int row  = blockIdx.x * 8 + wave;
    if (row >= n) return;

    float v[4];
    float s = 0.0f, s2 = 0.0f;
    #pragma unroll
    for (int j = 0; j < 4; ++j) {
        const int c = lane + 32 * j;
        const float r = q[row * DIMC + c] + P[row * DIMC + c];
        v[j] = r;
        s  += r;
        s2 += r * r;
    }
    #pragma unroll
    for (int off = 16; off > 0; off >>= 1) {        // wave32 tree reduce
        s  += __shfl_xor(s,  off, 32);
        s2 += __shfl_xor(s2, off, 32);
    }
    const float mu  = s * (1.0f / DIMC);
    const float var = s2 * (1.0f / DIMC) - mu * mu;
    const float inv = rsqrtf(var + 1e-5f);
    #pragma unroll
    for (int j = 0; j < 4; ++j) {
        const int c = lane + 32 * j;
        out[row * DIMC + c] = gamma[c] * (v[j] - mu) * inv + beta[c];
    }
}

// ---------------------------------------------------------------------------
extern "C" void kernel_launch(void* const* d_in, const int* in_sizes, int n_in,
                              void* d_out, int out_size, void* d_ws, size_t ws_size,
                              hipStream_t stream)
{
    const float* query = (const float*)d_in[0];
    const float* key   = (const float*)d_in[1];
    const float* value = (const float*)d_in[2];
    const int*   qidx  = (const int*)  d_in[3];
    const int*   kidx  = (const int*)  d_in[4];
    const float* Wq    = (const float*)d_in[5];
    const float* bq    = (const float*)d_in[6];
    const float* Wk    = (const float*)d_in[7];
    const float* bk    = (const float*)d_in[8];
    const float* Wv    = (const float*)d_in[9];
    const float* bv    = (const float*)d_in[10];
    const float* Wp    = (const float*)d_in[11];
    const float* bp    = (const float*)d_in[12];
    const float* a     = (const float*)d_in[13];
    const float* alpha = (const float*)d_in[14];
    const float* gamma = (const float*)d_in[15];
    const float* beta  = (const float*)d_in[16];
    float*       out   = (float*)d_out;

    const int N = in_sizes[0] / DIMC;
    const int M = in_sizes[1] / DIMC;
    const int E = in_sizes[3];

    // Workspace carve-up (floats), 256B-aligned base chunks.
    float* ws = (float*)d_ws;
    size_t off = 0;
    auto take = [&](size_t nfloats) {
        float* p = ws + off;
        off += (nfloats + 63) & ~(size_t)63;
        return p;
    };
    float* Qq      = take((size_t)N * DIMC);
    float* Kq      = take((size_t)N * DIMC);
    float* Kk      = take((size_t)M * DIMC);
    float* Vq      = take((size_t)N * DIMC);   // reused as P after softmax init
    float* Vv      = take((size_t)M * DIMC);
    float* attn_qq = take((size_t)N * HC);
    float* maxv    = take((size_t)N * HC);
    float* denom   = take((size_t)N * HC);
    float* attn_e  = take((size_t)E * HC);
    float* msg     = take((size_t)N * DIMC);
    (void)ws_size;

    const int gemmGridN = (N + 15) / 16;
    const int gemmGridM = (M + 15) / 16;

    // 1) five projections (WMMA f32 16x16x4)
    gat_gemm128_kernel<<<gemmGridN, 256, 0, stream>>>(query, Wq, bq, Qq, N);
    gat_gemm128_kernel<<<gemmGridN, 256, 0, stream>>>(query, Wk, bk, Kq, N);
    gat_gemm128_kernel<<<gemmGridM, 256, 0, stream>>>(key,   Wk, bk, Kk, M);
    gat_gemm128_kernel<<<gemmGridN, 256, 0, stream>>>(query, Wv, bv, Vq, N);
    gat_gemm128_kernel<<<gemmGridM, 256, 0, stream>>>(value, Wv, bv, Vv, M);

    // 2) node self-attention scores + max init
    const int nodeGrid = (N * HC + 255) / 256;
    gat_node_attn_kernel<<<nodeGrid, 256, 0, stream>>>(Qq, Kq, a, alpha,
                                                       attn_qq, maxv, N);
    // 3) edge scores + scatter max
    const int edgeGrid = (E * HC + 255) / 256;
    gat_edge_attn_kernel<<<edgeGrid, 256, 0, stream>>>(Qq, Kk, qidx, kidx, a,
                                                       alpha, attn_e, maxv, E);
    // 4) softmax init (denom, msg)
    gat_softmax_init_kernel<<<nodeGrid, 256, 0, stream>>>(attn_qq, maxv, Vq,
                                                          denom, msg, N);
    // 5) edge value scatter
    gat_edge_scatter_kernel<<<edgeGrid, 256, 0, stream>>>(attn_e, maxv, Vv,
                                                          qidx, kidx, denom,
                                                          msg, E);
    // 6) msg /= denom (in place)
    const int divGrid = (N * DIMC + 255) / 256;
    gat_div_kernel<<<divGrid, 256, 0, stream>>>(msg, denom, N);

    // 7) output projection into Vq buffer (dead after step 4)
    float* P = Vq;
    gat_gemm128_kernel<<<gemmGridN, 256, 0, stream>>>(msg, Wp, bp, P, N);

    // 8) residual + layernorm -> out
    const int lnGrid = (N + 7) / 8;
    gat_residual_ln_kernel<<<lnGrid, 256, 0, stream>>>(query, P, gamma, beta,
                                                       out, N);
}